// StaticSAGE_60790967108374
// MI455X (gfx1250) — compile-verified
//
#include <hip/hip_runtime.h>

#define HDIM 128
#define ASTRIDE 132   // padded LDS row stride (dwords): 132 % 64 = 4 -> conflict-free rows, 16B aligned

typedef __attribute__((ext_vector_type(2))) float v2f;
typedef __attribute__((ext_vector_type(8))) float v8f;

// ---------------------------------------------------------------------------
// h[n] = emb[node_ids[n]]  (float4 granularity, coalesced 128B per 8 lanes)
// ---------------------------------------------------------------------------
__global__ void gather_kernel(const float* __restrict__ emb,
                              const int* __restrict__ ids,
                              float* __restrict__ h, int nrows) {
    int i = blockIdx.x * blockDim.x + threadIdx.x;   // over nrows * 32 float4s
    if (i >= nrows * 32) return;
    int n = i >> 5;
    int q = i & 31;
    const float4* s = reinterpret_cast<const float4*>(emb + (size_t)ids[n] * HDIM);
    float4*       d = reinterpret_cast<float4*>(h + (size_t)n * HDIM);
    d[q] = s[q];
}

// ---------------------------------------------------------------------------
// out = relu(hin @ W + bias)   [N,128] x [128,128]
// 8 waves/block; wave w owns output columns [16w, 16w+16).
// B fragments (32 k-steps) live in VGPRs for the whole kernel; A strip in LDS.
// ---------------------------------------------------------------------------
__global__ void __launch_bounds__(256, 1)
pool_mlp_kernel(const float* __restrict__ hin, const float* __restrict__ W,
                const float* __restrict__ bias, float* __restrict__ out,
                int N, int nstrips) {
    __shared__ float Alds[16 * ASTRIDE];

    const int tid   = threadIdx.x;
    const int lane  = tid & 31;
    const int wave  = tid >> 5;          // 0..7
    const int cb    = wave * 16;         // column base of this wave's tile
    const int ln    = lane & 15;         // M (for A) / N (for B,D)
    const int khalf = (lane >> 4) * 2;   // K sub-offset within a k-step

    // Preload B: W[k][n] fragments for all 32 k-steps (read once per block from L2)
    v2f bfrag[32];
#pragma unroll
    for (int ks = 0; ks < 32; ++ks) {
        int k = ks * 4 + khalf;
        bfrag[ks].x = W[(size_t)k * HDIM + cb + ln];
        bfrag[ks].y = W[(size_t)(k + 1) * HDIM + cb + ln];
    }
    const float bv = bias[cb + ln];

    for (int strip = blockIdx.x; strip < nstrips; strip += gridDim.x) {
        const int rb = strip * 16;
        __syncthreads();   // protect previous iteration's LDS reads
        // Stage 16x128 A strip: 512 float4, 2 per thread, coalesced
#pragma unroll
        for (int j = tid; j < 512; j += 256) {
            int r  = j >> 5;
            int c4 = (j & 31) << 2;
            int rg = rb + r; if (rg >= N) rg = N - 1;
            float4 v = *reinterpret_cast<const float4*>(hin + (size_t)rg * HDIM + c4);
            *reinterpret_cast<float4*>(&Alds[r * ASTRIDE + c4]) = v;
        }
        __syncthreads();

        v8f c = {};
#pragma unroll
        for (int ks = 0; ks < 32; ++ks) {
            int k = ks * 4 + khalf;
            v2f a = *reinterpret_cast<const v2f*>(&Alds[ln * ASTRIDE + k]);
            c = __builtin_amdgcn_wmma_f32_16x16x4_f32(false, a, false, bfrag[ks],
                                                      (short)0, c, false, false);
        }
#pragma unroll
        for (int v = 0; v < 8; ++v) {
            int row = rb + v + (lane >> 4) * 8;
            if (row < N) {
                float val = c[v] + bv;
                out[(size_t)row * HDIM + cb + ln] = val > 0.0f ? val : 0.0f;
            }
        }
    }
}

// ---------------------------------------------------------------------------
// agg[n] = max over edges e with dst[e]==n of m[src[e]]; empty -> 0
// dst sorted; one block per node, 128 threads = feature lanes.
// Binary search is wave-uniform -> scalar; edge row reads are 512B coalesced.
// ---------------------------------------------------------------------------
__global__ void segmax_kernel(const float* __restrict__ m,
                              const int* __restrict__ src,
                              const int* __restrict__ dst,
                              float* __restrict__ agg, int E) {
    const int n = blockIdx.x;
    const int f = threadIdx.x;

    int lo = 0, hi = E;
    while (lo < hi) { int mid = (lo + hi) >> 1; if (dst[mid] <  n) lo = mid + 1; else hi = mid; }
    const int start = lo;
    hi = E;
    while (lo < hi) { int mid = (lo + hi) >> 1; if (dst[mid] <= n) lo = mid + 1; else hi = mid; }
    const int end = lo;

    float amax = -__builtin_inff();
    for (int e = start; e < end; ++e) {
        int s = src[e];                                   // uniform -> scalar load
        amax = fmaxf(amax, m[(size_t)s * HDIM + f]);
    }
    agg[(size_t)n * HDIM + f] = (end > start) ? amax : 0.0f;
}

// ---------------------------------------------------------------------------
// out = hin @ Wself + agg @ Wneigh + bias   (fused, two B sets in VGPRs)
// ---------------------------------------------------------------------------
__global__ void __launch_bounds__(256, 1)
out_gemm_kernel(const float* __restrict__ hin, const float* __restrict__ agg,
                const float* __restrict__ Wself, const float* __restrict__ Wneigh,
                const float* __restrict__ bias, float* __restrict__ out,
                int N, int nstrips) {
    __shared__ float Alds[2 * 16 * ASTRIDE];

    const int tid   = threadIdx.x;
    const int lane  = tid & 31;
    const int wave  = tid >> 5;
    const int cb    = wave * 16;
    const int ln    = lane & 15;
    const int khalf = (lane >> 4) * 2;

    v2f bS[32], bN[32];
#pragma unroll
    for (int ks = 0; ks < 32; ++ks) {
        int k = ks * 4 + khalf;
        bS[ks].x = Wself [(size_t)k * HDIM + cb + ln];
        bS[ks].y = Wself [(size_t)(k + 1) * HDIM + cb + ln];
        bN[ks].x = Wneigh[(size_t)k * HDIM + cb + ln];
        bN[ks].y = Wneigh[(size_t)(k + 1) * HDIM + cb + ln];
    }
    const float bv = bias[cb + ln];

    for (int strip = blockIdx.x; strip < nstrips; strip += gridDim.x) {
        const int rb = strip * 16;
        __syncthreads();
#pragma unroll
        for (int j = tid; j < 1024; j += 256) {          // 512 f4 of h, 512 f4 of agg
            int half = j >> 9;                           // 0 = h, 1 = agg
            int jj = j & 511;
            int r  = jj >> 5;
            int c4 = (jj & 31) << 2;
            int rg = rb + r; if (rg >= N) rg = N - 1;
            const float* srcp = half ? agg : hin;
            float4 v = *reinterpret_cast<const float4*>(srcp + (size_t)rg * HDIM + c4);
            *reinterpret_cast<float4*>(&Alds[(half * 16 + r) * ASTRIDE + c4]) = v;
        }
        __syncthreads();

        v8f c = {};
#pragma unroll
        for (int ks = 0; ks < 32; ++ks) {
            int k = ks * 4 + khalf;
            v2f a = *reinterpret_cast<const v2f*>(&Alds[ln * ASTRIDE + k]);
            c = __builtin_amdgcn_wmma_f32_16x16x4_f32(false, a, false, bS[ks],
                                                      (short)0, c, false, false);
        }
#pragma unroll
        for (int ks = 0; ks < 32; ++ks) {
            int k = ks * 4 + khalf;
            v2f a = *reinterpret_cast<const v2f*>(&Alds[(16 + ln) * ASTRIDE + k]);
            c = __builtin_amdgcn_wmma_f32_16x16x4_f32(false, a, false, bN[ks],
                                                      (short)0, c, false, false);
        }
#pragma unroll
        for (int v = 0; v < 8; ++v) {
            int row = rb + v + (lane >> 4) * 8;
            if (row < N)
                out[(size_t)row * HDIM + cb + ln] = c[v] + bv;
        }
    }
}

// ---------------------------------------------------------------------------
extern "C" void kernel_launch(void* const* d_in, const int* in_sizes, int n_in,
                              void* d_out, int out_size, void* d_ws, size_t ws_size,
                              hipStream_t stream) {
    const int*   node_ids = (const int*)  d_in[0];
    const int*   src      = (const int*)  d_in[1];
    const int*   dst      = (const int*)  d_in[2];
    const float* emb      = (const float*)d_in[3];
    const float* Wp1      = (const float*)d_in[4];
    const float* bp1      = (const float*)d_in[5];
    const float* Ws1      = (const float*)d_in[6];
    const float* Wn1      = (const float*)d_in[7];
    const float* b1       = (const float*)d_in[8];
    const float* Wp2      = (const float*)d_in[9];
    const float* bp2      = (const float*)d_in[10];
    const float* Ws2      = (const float*)d_in[11];
    const float* Wn2      = (const float*)d_in[12];
    const float* b2       = (const float*)d_in[13];

    const int N = in_sizes[0];
    const int E = in_sizes[1];
    float* out = (float*)d_out;

    const size_t NH = (size_t)N * HDIM;
    float* buf0 = (float*)d_ws;        // h0  -> m2
    float* buf1 = buf0 + NH;           // m1  -> h1
    float* buf2 = buf1 + NH;           // agg

    const int nstrips = (N + 15) / 16;
    const int GEMM_BLOCKS = 512;

    // h0 = emb[node_ids]
    gather_kernel<<<(N * 32 + 255) / 256, 256, 0, stream>>>(emb, node_ids, buf0, N);

    // ---- layer 1 ----
    pool_mlp_kernel<<<GEMM_BLOCKS, 256, 0, stream>>>(buf0, Wp1, bp1, buf1, N, nstrips);
    segmax_kernel<<<N, 128, 0, stream>>>(buf1, src, dst, buf2, E);
    out_gemm_kernel<<<GEMM_BLOCKS, 256, 0, stream>>>(buf0, buf2, Ws1, Wn1, b1, buf1, N, nstrips);

    // ---- layer 2 ----
    pool_mlp_kernel<<<GEMM_BLOCKS, 256, 0, stream>>>(buf1, Wp2, bp2, buf0, N, nstrips);
    segmax_kernel<<<N, 128, 0, stream>>>(buf0, src, dst, buf2, E);
    out_gemm_kernel<<<GEMM_BLOCKS, 256, 0, stream>>>(buf1, buf2, Ws2, Wn2, b2, out, N, nstrips);
}